// PatchTokenizer_3384434229558
// MI455X (gfx1250) — compile-verified
//
#include <hip/hip_runtime.h>
#include <stdint.h>

// ---------------------------------------------------------------------------
// PatchTokenizer for MI455X (gfx1250).
// Memory-bound scan/gather workload: no matrix ops, so no WMMA; instead we use
// the CDNA5 async global->LDS path (ASYNCcnt) to stage histogram tiles, and
// LDS atomics for the 512-bin histograms. Input (100MB) fits in 192MB L2, so
// the gather passes hit L2.
// ---------------------------------------------------------------------------

#define B_      32
#define IMG_    512
#define NC_     8192        // coarse patches total (B*16*16)
#define NF_     32768       // fine patches total (B*32*32)

__device__ __forceinline__ void async_load_b128(uint32_t lds_off, uint64_t gaddr) {
    asm volatile("global_load_async_to_lds_b128 %0, %1, off"
                 :: "v"(lds_off), "v"(gaddr) : "memory");
}
__device__ __forceinline__ void wait_asynccnt0() {
    asm volatile("s_wait_asynccnt 0" ::: "memory");
}

// -------------------- Kernel 1: entropy -> coarse mask ---------------------
// one 256-thread block per 32x32 coarse patch
__global__ void entropy_kernel(const float* __restrict__ img,
                               const float* __restrict__ mean,
                               const float* __restrict__ std_,
                               int* __restrict__ m32) {
    __shared__ float simg[3 * 32 * 32];   // 12KB staged tile (3 channels)
    __shared__ int   hist[512];
    __shared__ float red[256];

    const int tid = threadIdx.x;
    const int p   = blockIdx.x;            // coarse patch id
    const int b   = p >> 8;
    const int ci  = (p >> 4) & 15;
    const int cj  = p & 15;

    // Stage 768 float4's (3ch * 32 rows * 8 xfers) via async LDS loads.
#pragma unroll
    for (int k = 0; k < 3; ++k) {
        int q   = tid + (k << 8);          // [0,768)
        int c   = q >> 8;
        int rem = q & 255;
        int y   = rem >> 3;
        int x4  = rem & 7;
        const float* src = img + ((size_t)(b * 3 + c) * IMG_ + (ci * 32 + y)) * IMG_
                               + (cj * 32 + x4 * 8);
        uint32_t lds = (uint32_t)(uintptr_t)&simg[(c * 32 + y) * 32 + x4 * 8];
        async_load_b128(lds, (uint64_t)(uintptr_t)src);
    }
    hist[tid]       = 0;
    hist[tid + 256] = 0;
    wait_asynccnt0();          // our own asyncs done
    __syncthreads();           // everyone's asyncs + hist zero visible

    const float mn0 = mean[0], mn1 = mean[1], mn2 = mean[2];
    const float sd0 = std_[0], sd1 = std_[1], sd2 = std_[2];

#pragma unroll
    for (int k = 0; k < 4; ++k) {
        int idx = tid + (k << 8);          // pixel in [0,1024)
        int y = idx >> 5, x = idx & 31;
        float r = simg[(0 * 32 + y) * 32 + x];
        float g = simg[(1 * 32 + y) * 32 + x];
        float bl = simg[(2 * 32 + y) * 32 + x];
        float ur = fminf(fmaxf((r  * sd0 + mn0) * 255.0f, 0.0f), 255.0f);
        float ug = fminf(fmaxf((g  * sd1 + mn1) * 255.0f, 0.0f), 255.0f);
        float ub = fminf(fmaxf((bl * sd2 + mn2) * 255.0f, 0.0f), 255.0f);
        float gray = 0.2989f * ur + 0.587f * ug + 0.114f * ub;
        int bin = (int)(gray * 2.0f);      // bins/256 == 2.0, trunc like astype(int32)
        bin = min(max(bin, 0), 511);
        atomicAdd(&hist[bin], 1);          // ds_add_u32
    }
    __syncthreads();

    float local = 0.0f;
#pragma unroll
    for (int k = 0; k < 2; ++k) {
        int cnt = hist[tid + (k << 8)];
        float pv = (float)cnt * (1.0f / 1024.0f);
        local += pv * log2f(pv + 1e-10f);
    }
    red[tid] = -local;
    __syncthreads();
    for (int off = 128; off > 0; off >>= 1) {
        if (tid < off) red[tid] += red[tid + off];
        __syncthreads();
    }
    if (tid == 0) m32[p] = (red[0] < 7.2f) ? 1 : 0;
}

// ------------------ Kernel 2: scans + scalar tail outputs ------------------
// meta[0]=N32, meta[1]=N16, meta[2..34]=cu_seqlens[33]
__device__ __forceinline__ int fine_parent(int f) {
    int b = f >> 10, rem = f & 1023, r = rem >> 5, c = rem & 31;
    return (b << 8) | ((r >> 1) << 4) | (c >> 1);
}

__global__ void scan_kernel(const int* __restrict__ m32,
                            int* __restrict__ cpref,
                            int* __restrict__ fpref,
                            int* __restrict__ meta,
                            float* __restrict__ out) {
    __shared__ int tsum[1024];
    __shared__ int seql[32];
    __shared__ int cu[33];
    const int tid = threadIdx.x;

    // ---- phase 1: coarse exclusive scan (8 per thread) ----
    int base = tid * 8, s = 0;
#pragma unroll
    for (int k = 0; k < 8; ++k) s += m32[base + k];
    tsum[tid] = s;
    __syncthreads();
    for (int off = 1; off < 1024; off <<= 1) {
        int v = tsum[tid];
        int a = (tid >= off) ? tsum[tid - off] : 0;
        __syncthreads();
        tsum[tid] = v + a;
        __syncthreads();
    }
    int run = tsum[tid] - s;
#pragma unroll
    for (int k = 0; k < 8; ++k) { cpref[base + k] = run; run += m32[base + k]; }
    const int N32 = tsum[1023];

    if (tid < 32) {                         // per-batch coarse counts (32 thr x 8 = 256/batch)
        int hi = tsum[32 * tid + 31];
        int lo = tid ? tsum[32 * tid - 1] : 0;
        int n32b = hi - lo;
        seql[tid] = 1025 - 3 * n32b;        // 1 + 4*(256-n32b) + n32b
    }
    __syncthreads();
    if (tid == 0) {
        cu[0] = 0;
        for (int b = 0; b < 32; ++b) cu[b + 1] = cu[b] + seql[b];
        meta[0] = N32;
        for (int i = 0; i < 33; ++i) meta[2 + i] = cu[i];
    }
    __syncthreads();                        // all tsum reads done before reuse

    // ---- phase 2: fine exclusive scan (32 per thread) ----
    int fbase = tid * 32; s = 0;
#pragma unroll
    for (int k = 0; k < 32; ++k) s += 1 - m32[fine_parent(fbase + k)];
    tsum[tid] = s;
    __syncthreads();
    for (int off = 1; off < 1024; off <<= 1) {
        int v = tsum[tid];
        int a = (tid >= off) ? tsum[tid - off] : 0;
        __syncthreads();
        tsum[tid] = v + a;
        __syncthreads();
    }
    run = tsum[tid] - s;
#pragma unroll
    for (int k = 0; k < 32; ++k) {
        int f = fbase + k;
        int flag = 1 - m32[fine_parent(f)];
        fpref[f] = run; run += flag;
    }
    const int N16 = tsum[1023];
    if (tid == 0) {
        meta[1] = N16;
        // scalar tail outputs
        long long off_flat16  = (long long)N16 * 768 + (long long)N32 * 3840;
        long long off_outmask = off_flat16 + 32768 + 8192;
        int total = cu[32];                 // 32 + N16 + N32
        long long off_seq = off_outmask + total;
        for (int b = 0; b < 32; ++b) out[off_seq + b] = (float)seql[b];
        for (int i = 0; i < 33; ++i) out[off_seq + 32 + i] = (float)cu[i];
        int mx = 0;
        for (int b = 0; b < 32; ++b) mx = max(mx, seql[b]);
        out[off_seq + 65] = (float)mx;
        out[off_seq + 66] = (float)total / 32768.0f;   // / (B*H*W/PS^2)
    }
}

// ------------- Kernel 3: flat16/flat32 bools + output_mask -----------------
__global__ void tail_kernel(const int* __restrict__ m32,
                            const int* __restrict__ meta,
                            float* __restrict__ out) {
    int idx = blockIdx.x * blockDim.x + threadIdx.x;
    const int N32 = meta[0], N16 = meta[1];
    long long off_flat16  = (long long)N16 * 768 + (long long)N32 * 3840;
    long long off_flat32  = off_flat16 + 32768;
    long long off_outmask = off_flat32 + 8192;

    if (idx < NF_) {
        out[off_flat16 + idx] = m32[fine_parent(idx)] ? 0.0f : 1.0f;
    } else if (idx < NF_ + NC_) {
        int j = idx - NF_;
        out[off_flat32 + j] = m32[j] ? 1.0f : 0.0f;
    } else {
        int j = idx - (NF_ + NC_);
        int total = meta[2 + 32];
        if (j < total) {
            int b = 0;
            for (int k = 1; k < 32; ++k) if (j >= meta[2 + k]) b = k;
            int r    = j - meta[2 + b];
            int seq  = meta[2 + b + 1] - meta[2 + b];
            int n32b = (1025 - seq) / 3;
            int n16b = seq - 1 - n32b;
            out[off_outmask + j] = (r == 0) ? -1.0f : (r <= n16b ? 1.0f : 2.0f);
        }
    }
}

// ----------------- Kernel 4: gather kept 16x16 fine patches ----------------
// 192 threads/block: one float4 each (3ch * 16 rows * 4 xfers)
__global__ void gather16_kernel(const float* __restrict__ img,
                                const int* __restrict__ m32,
                                const int* __restrict__ fpref,
                                float* __restrict__ out) {
    int p = blockIdx.x;                     // fine patch id
    int b = p >> 10, rem = p & 1023, r = rem >> 5, c = rem & 31;
    if (m32[(b << 8) | ((r >> 1) << 4) | (c >> 1)]) return;  // parent kept -> fine dropped
    long long dst = (long long)fpref[p] * 768;
    int t = threadIdx.x;
    int ch = t >> 6, rem2 = t & 63, y = rem2 >> 2, x4 = rem2 & 3;
    const float4* src = (const float4*)(img + ((size_t)(b * 3 + ch) * IMG_ + (r * 16 + y)) * IMG_
                                            + c * 16 + x4 * 4);
    *(float4*)(out + dst + ch * 256 + y * 16 + x4 * 4) = *src;
}

// ------- Kernel 5: kept coarse patches -> full32 sub-patches + sel32 -------
__global__ void gather32_kernel(const float* __restrict__ img,
                                const int* __restrict__ m32,
                                const int* __restrict__ cpref,
                                const int* __restrict__ meta,
                                float* __restrict__ out) {
    int p = blockIdx.x;
    if (!m32[p]) return;
    int b = p >> 8, ci = (p >> 4) & 15, cj = p & 15;
    long long off_full32 = (long long)meta[1] * 768;
    long long off_sel32  = off_full32 + (long long)meta[0] * 3072;
    long long rr = cpref[p];
    int t = threadIdx.x;                    // 256

    // full32: [4][3][16][16] = 768 float4, 3 per thread
#pragma unroll
    for (int k = 0; k < 3; ++k) {
        int q = t + (k << 8);               // [0,768)
        int s = q / 192, rem = q % 192;
        int ch = rem >> 6, rem2 = rem & 63, y = rem2 >> 2, x4 = rem2 & 3;
        int di = s >> 1, dj = s & 1;
        const float4* src = (const float4*)(img + ((size_t)(b * 3 + ch) * IMG_
                                + (ci * 32 + di * 16 + y)) * IMG_
                                + cj * 32 + dj * 16 + x4 * 4);
        *(float4*)(out + off_full32 + rr * 3072 + s * 768 + ch * 256 + y * 16 + x4 * 4) = *src;
    }
    // sel32: 2x bilinear (antialias=False, exact 2x -> 2x2 mean), 3 floats/thread
#pragma unroll
    for (int k = 0; k < 3; ++k) {
        int q = t + (k << 8);               // [0,768)
        int ch = q >> 8, rem = q & 255, y = rem >> 4, x = rem & 15;
        const float* base = img + ((size_t)(b * 3 + ch) * IMG_ + (ci * 32 + 2 * y)) * IMG_
                                + cj * 32 + 2 * x;
        float2 a  = *(const float2*)base;
        float2 bb = *(const float2*)(base + IMG_);
        out[off_sel32 + rr * 768 + q] = (a.x + a.y + bb.x + bb.y) * 0.25f;
    }
}

// ---------------------------------------------------------------------------
extern "C" void kernel_launch(void* const* d_in, const int* in_sizes, int n_in,
                              void* d_out, int out_size, void* d_ws, size_t ws_size,
                              hipStream_t stream) {
    const float* images = (const float*)d_in[0];
    const float* mean   = (const float*)d_in[1];
    const float* std_   = (const float*)d_in[2];
    float* out = (float*)d_out;

    int* ws    = (int*)d_ws;
    int* meta  = ws;                 // 64 ints
    int* m32f  = ws + 64;            // 8192
    int* cpref = ws + 64 + NC_;      // 8192
    int* fpref = ws + 64 + 2 * NC_;  // 32768

    entropy_kernel<<<dim3(NC_), dim3(256), 0, stream>>>(images, mean, std_, m32f);
    scan_kernel<<<dim3(1), dim3(1024), 0, stream>>>(m32f, cpref, fpref, meta, out);
    // 32768 + 8192 + worst-case outmask (32 + 32768) = 73760 -> 290 blocks
    tail_kernel<<<dim3(290), dim3(256), 0, stream>>>(m32f, meta, out);
    gather16_kernel<<<dim3(NF_), dim3(192), 0, stream>>>(images, m32f, fpref, out);
    gather32_kernel<<<dim3(NC_), dim3(256), 0, stream>>>(images, m32f, cpref, meta, out);
}